// TableBatchedEmbeddingBags_23751169147036
// MI455X (gfx1250) — compile-verified
//
#include <hip/hip_runtime.h>
#include <stdint.h>

// Table-batched EmbeddingBag (sum pooling) for MI455X / gfx1250.
//
// Mapping: one wave32 per bag. Lane l owns dims [4l, 4l+4) of the 128-dim row,
// so one b128 op per lane x 32 lanes = one full 512B embedding row.
// Gathers are issued as async DMA into an 8-slot LDS ring (4 in flight),
// tracked with ASYNCcnt, and accumulated from LDS in f32.

#define WAVES_PER_BLOCK 8      // 256 threads = 8 wave32
#define RING            8      // LDS ring slots per wave
#define INFLIGHT        4      // async DMAs in flight per wave
#define D_DIM           128    // embedding dim (benchmark-fixed)
#define ROW_BYTES       512    // D_DIM * sizeof(float)

// True ext-vector (not HIP's struct float4) so it can be fed to inline asm
// with a "v" constraint as a direct 4-VGPR tuple.
typedef float v4f __attribute__((ext_vector_type(4)));

__global__ __launch_bounds__(WAVES_PER_BLOCK * 32)
void embbag_async_kernel(const float* __restrict__ weights,
                         const int*  __restrict__ table_offsets,
                         const int*  __restrict__ indices,
                         const int*  __restrict__ offsets,
                         float* __restrict__ out,
                         int n_bags, int batch, int batch_shift, int n_tables)
{
    __shared__ __align__(16) float lds_buf[WAVES_PER_BLOCK][RING][D_DIM];

    const int lane = threadIdx.x & 31;
    const int wave = threadIdx.x >> 5;
    const int seg  = blockIdx.x * WAVES_PER_BLOCK + wave;
    if (seg >= n_bags) return;

    // table id / batch id (batch is pow2 in the benchmark -> shift)
    const int t = (batch_shift >= 0) ? (seg >> batch_shift) : (seg / batch);
    const int b = seg - t * batch;

    const int start = offsets[seg];
    const int end   = offsets[seg + 1];
    const uint32_t tbase = (uint32_t)table_offsets[t];

    // Wave-relative LDS byte offsets for this wave's ring slots.
    // (low 32 bits of a generic pointer to __shared__ data == LDS offset)
    uint32_t lds_addr[RING];
#pragma unroll
    for (int s = 0; s < RING; ++s)
        lds_addr[s] = (uint32_t)(uintptr_t)(&lds_buf[wave][s][lane * 4]);

    const uint32_t lane_byte = (uint32_t)lane * 16u;

    v4f acc = {0.f, 0.f, 0.f, 0.f};

    // Process bag indices in chunks of <=32 so one coalesced b32 load grabs
    // the whole chunk of indices; broadcast each with v_readlane.
    for (int cbase = start; cbase < end; cbase += 32) {
        const int cnt = min(32, end - cbase);

        int myrow = 0;
        if (lane < cnt) myrow = indices[cbase + lane];

        // ---- prime the async pipeline (up to INFLIGHT DMAs) ----
        int issued = 0;
        const int prime = (cnt < INFLIGHT) ? cnt : INFLIGHT;
        for (; issued < prime; ++issued) {
            uint32_t row  = tbase + (uint32_t)__builtin_amdgcn_readlane(myrow, issued);
            uint32_t goff = row * (uint32_t)ROW_BYTES + lane_byte;
            asm volatile("global_load_async_to_lds_b128 %0, %1, %2"
                         :: "v"(lds_addr[issued & (RING - 1)]), "v"(goff), "s"(weights)
                         : "memory");
        }

        // ---- steady state: consume slot i, refill slot i+INFLIGHT ----
        // Refill target was consumed INFLIGHT iterations ago; the in-order DS
        // completion implied by consuming v_i already covers that read, so no
        // explicit dscnt drain is needed.
        int i = 0;
        for (; i + INFLIGHT < cnt; ++i) {
            // issued == i + INFLIGHT here; asynccnt <= INFLIGHT-1 => stage i done
            asm volatile("s_wait_asynccnt 3" ::: "memory");
            const v4f v = *(const v4f*)&lds_buf[wave][i & (RING - 1)][lane * 4];
            acc += v;
            uint32_t row  = tbase + (uint32_t)__builtin_amdgcn_readlane(myrow, issued);
            uint32_t goff = row * (uint32_t)ROW_BYTES + lane_byte;
            asm volatile("global_load_async_to_lds_b128 %0, %1, %2"
                         :: "v"(lds_addr[issued & (RING - 1)]), "v"(goff), "s"(weights)
                         : "memory");
            ++issued;
        }

        // ---- drain ----
        asm volatile("s_wait_asynccnt 0" ::: "memory");
        for (; i < cnt; ++i) {
            const v4f v = *(const v4f*)&lds_buf[wave][i & (RING - 1)][lane * 4];
            acc += v;
        }
    }

    // out[b][t][d]; streamed output, never re-read -> non-temporal store so it
    // doesn't evict embedding rows from L2 (random gathers have ~30% re-touch).
    float* op = out + ((size_t)b * n_tables + t) * D_DIM + lane * 4;
    asm volatile("global_store_b128 %0, %1, off th:TH_STORE_NT"
                 :: "v"(op), "v"(acc) : "memory");
}

extern "C" void kernel_launch(void* const* d_in, const int* in_sizes, int n_in,
                              void* d_out, int out_size, void* d_ws, size_t ws_size,
                              hipStream_t stream) {
    const float* weights       = (const float*)d_in[0];
    const int*   table_offsets = (const int*)d_in[1];
    const int*   indices       = (const int*)d_in[2];
    const int*   offsets       = (const int*)d_in[3];
    float*       out           = (float*)d_out;

    const int n_tables = in_sizes[1];
    const int n_bags   = in_sizes[3] - 1;
    const int batch    = n_bags / n_tables;
    // batch is a power of two in the benchmark; precompute the shift.
    const int batch_shift =
        (batch > 0 && (batch & (batch - 1)) == 0) ? __builtin_ctz(batch) : -1;

    const int blocks = (n_bags + WAVES_PER_BLOCK - 1) / WAVES_PER_BLOCK;
    embbag_async_kernel<<<blocks, WAVES_PER_BLOCK * 32, 0, stream>>>(
        weights, table_offsets, indices, offsets, out,
        n_bags, batch, batch_shift, n_tables);
}